// EnhancedMemoryManager_75110388073137
// MI455X (gfx1250) — compile-verified
//
#include <hip/hip_runtime.h>
#include <hip/hip_bf16.h>

// ---------------------------------------------------------------------------
// EnhancedMemoryManager update for MI455X (gfx1250, wave32).
// Memory-bound (~1.16 GB traffic -> ~50us at 23.3 TB/s). f32 throughout.
// CDNA5 paths used:
//   * V_WMMA_F32_16X16X4_F32 for the masked history reduction (per-user
//     mask[1xS] x feat[SxD] expressed as A=feat^T tile, B=mask broadcast).
//   * LDS staging (ds ops) for the feature tile.
//   * global_prefetch_b8 via __builtin_prefetch in the streaming copy.
// ---------------------------------------------------------------------------

typedef float v2f __attribute__((ext_vector_type(2)));
typedef float v8f __attribute__((ext_vector_type(8)));

#define MOM 0.9f
#define EPS_MASK 1e-6f
#define EPS_NORM 1e-12f

constexpr int P = 16;      // prototypes per user
constexpr int D = 128;     // embedding dim
constexpr int S = 50;      // history length
constexpr int SPAD = 52;   // S padded to multiple of 4 (zero fill)

// ---- bulk f32 copy (prototype table passthrough), float4 + prefetch -------
__global__ void copy_f4_kernel(const float4* __restrict__ src,
                               float4* __restrict__ dst, long n4) {
  long i = (long)blockIdx.x * blockDim.x + threadIdx.x;
  long stride = (long)gridDim.x * blockDim.x;
  for (; i < n4; i += stride) {
    __builtin_prefetch(src + i + stride, 0, 1);  // global_prefetch_b8
    dst[i] = src[i];
  }
}

__global__ void copy_f_kernel(const float* __restrict__ src,
                              float* __restrict__ dst, int n) {
  int i = blockIdx.x * blockDim.x + threadIdx.x;
  if (i < n) dst[i] = src[i];
}

// ---- per-user: masked reduction (WMMA), normalize, blend, scatter ---------
__global__ __launch_bounds__(256) void user_update_kernel(
    const float* __restrict__ protos,        // [U,P,D]
    const float* __restrict__ ic,            // [U]
    const float* __restrict__ feat,          // [B,S,D]
    const float* __restrict__ mask,          // [B,S]
    const int* __restrict__ uidx,            // [B]
    float* __restrict__ out_protos,          // [U,P,D] (pre-copied)
    float* __restrict__ out_ic,              // [U]     (pre-copied)
    float* __restrict__ ws_upd)              // [B,D] normalized upd
{
  __shared__ float feat_lds[SPAD * D];       // 26 KB, zero-padded rows
  __shared__ float mask_lds[SPAD];
  __shared__ float upd_lds[D];
  __shared__ float s_msum, s_inorm;

  const int b = blockIdx.x;
  const int tid = threadIdx.x;

  // stage features [S,D] into LDS, coalesced; zero-pad rows S..SPAD
  for (int i = tid; i < SPAD * D; i += 256)
    feat_lds[i] = (i < S * D) ? feat[(size_t)b * S * D + i] : 0.0f;
  if (tid < SPAD)
    mask_lds[tid] = (tid < S) ? mask[(size_t)b * S + tid] : 0.0f;
  __syncthreads();

  // WMMA f32 16x16x4 masked reduction.
  //   A[m,k] = feat[s0+k, d0+m]  (A layout: lanes 0-15 -> K0,K1; 16-31 -> K2,K3)
  //   B[k,n] = mask[s0+k]        (independent of n)
  //   D[m,n] = sum_k feat[s0+k, d0+m]*mask[s0+k]  -> partial upd for dim d0+m
  const int lane = tid & 31;
  const int wave = tid >> 5;          // 8 waves x 16 dims = 128 dims
  const int hl   = lane >> 4;         // lane half selects K pair
  const int m16  = lane & 15;         // M index within tile
  const int d0   = wave * 16;

  v8f c = {0.f, 0.f, 0.f, 0.f, 0.f, 0.f, 0.f, 0.f};
#pragma unroll
  for (int s0 = 0; s0 < SPAD; s0 += 4) {
    const int k0 = s0 + 2 * hl;
    v2f a, bb;
    a.x  = feat_lds[(k0    ) * D + d0 + m16];
    a.y  = feat_lds[(k0 + 1) * D + d0 + m16];
    bb.x = mask_lds[k0];
    bb.y = mask_lds[k0 + 1];
    c = __builtin_amdgcn_wmma_f32_16x16x4_f32(
        /*neg_a=*/false, a, /*neg_b=*/false, bb,
        /*c_mod=*/(short)0, c, /*reuse_a=*/false, /*reuse_b=*/false);
  }
  // D extraction: VGPR v on lane L holds D[v + 8*(L>=16)][L&15]; columns equal.
  if (m16 == 0) {
#pragma unroll
    for (int v = 0; v < 8; ++v) upd_lds[d0 + v + 8 * hl] = c[v];
  }
  if (tid == 0) {
    float s = 0.0f;
    for (int i = 0; i < S; ++i) s += mask_lds[i];
    s_msum = fmaxf(s, EPS_MASK);
  }
  __syncthreads();

  if (tid < D) upd_lds[tid] = upd_lds[tid] / s_msum;
  __syncthreads();
  if (tid == 0) {
    float n = 0.0f;
    for (int i = 0; i < D; ++i) n += upd_lds[i] * upd_lds[i];
    s_inorm = 1.0f / fmaxf(sqrtf(n), EPS_NORM);
  }
  __syncthreads();
  if (tid < D) {
    float u = upd_lds[tid] * s_inorm;
    upd_lds[tid] = u;
    ws_upd[(size_t)b * D + tid] = u;   // for shared-prototype mean
  }
  __syncthreads();

  // momentum blend + scatter write of this user's 16x128 prototype block
  const int uid = uidx[b];
  const float m = fminf(fmaxf(MOM + ic[uid] * (1.0f / 1000.0f), MOM), 0.99f);
  const size_t base = (size_t)uid * P * D;
  for (int i = tid; i < P * D; i += 256) {
    const float cur = protos[base + i];
    out_protos[base + i] = m * cur + (1.0f - m) * upd_lds[i & (D - 1)];
  }
  if (tid == 0) atomicAdd(&out_ic[uid], 1.0f);  // unique idx -> deterministic
}

// ---- shared prototype: deterministic two-stage mean over B ----------------
__global__ __launch_bounds__(128) void shared_sum_stage1(
    const float* __restrict__ ws_upd, float* __restrict__ partial, int B) {
  const int g = blockIdx.x;                 // fixed 128 blocks
  const int d = threadIdx.x;
  const int chunk = (B + gridDim.x - 1) / gridDim.x;
  const int b0 = g * chunk;
  const int b1 = min(B, b0 + chunk);
  float s = 0.0f;
  for (int b = b0; b < b1; ++b) s += ws_upd[(size_t)b * D + d];
  partial[g * D + d] = s;
}

__global__ __launch_bounds__(128) void shared_finish_kernel(
    const float* __restrict__ partial, int nparts, int B,
    const float* __restrict__ shared_protos, float* __restrict__ out_shared) {
  __shared__ float mean_lds[D];
  __shared__ float s_inorm;
  const int d = threadIdx.x;
  float s = 0.0f;
  for (int g = 0; g < nparts; ++g) s += partial[g * D + d];
  mean_lds[d] = s / (float)B;
  __syncthreads();
  if (d == 0) {
    float n = 0.0f;
    for (int i = 0; i < D; ++i) n += mean_lds[i] * mean_lds[i];
    s_inorm = 1.0f / fmaxf(sqrtf(n), EPS_NORM);
  }
  __syncthreads();
  const float sh = mean_lds[d] * s_inorm;
  for (int p = 0; p < P; ++p)
    out_shared[p * D + d] =
        MOM * shared_protos[p * D + d] + (1.0f - MOM) * sh;
}

// ---------------------------------------------------------------------------
extern "C" void kernel_launch(void* const* d_in, const int* in_sizes, int n_in,
                              void* d_out, int out_size, void* d_ws,
                              size_t ws_size, hipStream_t stream) {
  const float* protos        = (const float*)d_in[0];  // [U,P,D]
  const float* shared_protos = (const float*)d_in[1];  // [1,P,D]
  const float* ic            = (const float*)d_in[2];  // [U]
  const float* feat          = (const float*)d_in[3];  // [B,S,D]
  const float* mask          = (const float*)d_in[4];  // [B,S]
  const int*   uidx          = (const int*)d_in[5];    // [B]

  const int U = in_sizes[2];
  const int B = in_sizes[5];

  float* out         = (float*)d_out;
  float* out_protos  = out;                              // U*P*D
  float* out_shared  = out + (size_t)U * P * D;          // P*D
  float* out_ic      = out_shared + P * D;               // U

  float* ws_upd     = (float*)d_ws;                      // B*D floats
  float* ws_partial = ws_upd + (size_t)B * D;            // 128*D floats

  // 1) passthrough copies (out = in), then overwritten at user_idx rows
  const long n4 = ((long)U * P * D) / 4;
  copy_f4_kernel<<<2048, 256, 0, stream>>>(
      (const float4*)protos, (float4*)out_protos, n4);
  copy_f_kernel<<<(U + 255) / 256, 256, 0, stream>>>(ic, out_ic, U);

  // 2) per-user masked reduction (WMMA), normalize, blend, scatter
  user_update_kernel<<<B, 256, 0, stream>>>(
      protos, ic, feat, mask, uidx, out_protos, out_ic, ws_upd);

  // 3) shared prototype update (deterministic tree sum over B)
  shared_sum_stage1<<<128, 128, 0, stream>>>(ws_upd, ws_partial, B);
  shared_finish_kernel<<<1, 128, 0, stream>>>(ws_partial, 128, B,
                                              shared_protos, out_shared);
}